// LayerScale_25125558681902
// MI455X (gfx1250) — compile-verified
//
#include <hip/hip_runtime.h>
#include <stdint.h>

// LayerScale: out[b,s,d] = in[b,s,d] * gamma[d]
// B=4, S=8192, D=1024, fp32. Pure HBM-stream (0.125 FLOP/byte) -> the optimal
// MI455X kernel is 128-bit non-temporal traffic + async-to-LDS gamma staging.
// WMMA is deliberately absent: recasting as diag-matmul adds 2048x FLOPs for
// bit-identical output and cannot beat the 23.3 TB/s memory floor (~11.5 us).

typedef float v4f __attribute__((ext_vector_type(4)));

#define LS_D 1024          // channel dim (fixed by the reference)
#define LS_BLOCK 256       // 8 wave32s; 256 threads * float4 == one row

__global__ __launch_bounds__(LS_BLOCK) void layerscale_kernel(
    const float* __restrict__ in,
    const float* __restrict__ gamma,
    float* __restrict__ out,
    int n_rows)
{
    __shared__ float sgam[LS_D];
    const int tid = threadIdx.x;           // 0..255
    const int col = tid * 4;               // this thread's 4 channels

    // ---- CDNA5 async copy: gamma (4 KB) global -> LDS -------------------
    // One global_load_async_to_lds_b128 per lane: 256 lanes x 16B = whole
    // gamma vector in a single instruction, tracked by ASYNCcnt.
    {
        uint32_t lds_off = (uint32_t)(uintptr_t)(&sgam[col]); // low 32b = LDS byte offset
        const float* gsrc = gamma + col;
        asm volatile(
            "global_load_async_to_lds_b128 %0, %1, off\n\t"
            "s_wait_asynccnt 0"
            :
            : "v"(lds_off), "v"(gsrc)
            : "memory");
    }
    __syncthreads();

    // This thread's 4 gamma values live in registers for the whole kernel
    // (gamma slice is loop-invariant because blockDim*4 == D exactly).
    const v4f g = *(const v4f*)(&sgam[col]);

    // ---- Streaming loop: pointer-increment, 2x unrolled -----------------
    // Per trip: two b128 NT loads issued back-to-back (2x MLP), 8 v_mul_f32,
    // two b128 NT stores; addresses advance by a loop-invariant stride.
    const size_t step = (size_t)gridDim.x * LS_D;   // elems between my rows
    size_t off = (size_t)blockIdx.x * LS_D + (size_t)col;

    int row = blockIdx.x;
    const int gsz = gridDim.x;

    for (; row + gsz < n_rows; row += 2 * gsz) {
        v4f x0 = __builtin_nontemporal_load((const v4f*)(in + off));
        v4f x1 = __builtin_nontemporal_load((const v4f*)(in + off + step));
        __builtin_nontemporal_store(x0 * g, (v4f*)(out + off));
        __builtin_nontemporal_store(x1 * g, (v4f*)(out + off + step));
        off += 2 * step;
    }
    if (row < n_rows) {   // tail (not taken for B*S=32768 with grid=2048)
        v4f x = __builtin_nontemporal_load((const v4f*)(in + off));
        __builtin_nontemporal_store(x * g, (v4f*)(out + off));
    }
}

extern "C" void kernel_launch(void* const* d_in, const int* in_sizes, int n_in,
                              void* d_out, int out_size, void* d_ws, size_t ws_size,
                              hipStream_t stream) {
    (void)n_in; (void)d_ws; (void)ws_size; (void)out_size;

    const float* in    = (const float*)d_in[0];   // (B,S,D) fp32
    const float* gamma = (const float*)d_in[1];   // (D,)    fp32
    float* out         = (float*)d_out;

    const int n_rows = in_sizes[0] / LS_D;        // B*S = 32768 rows

    // 2048 persistent blocks: 16 rows each (8 clean 2x iterations),
    // 2048 x 8 wave32 = 16K waves in flight -> saturates 23.3 TB/s.
    int grid = n_rows < 2048 ? n_rows : 2048;

    layerscale_kernel<<<grid, LS_BLOCK, 0, stream>>>(in, gamma, out, n_rows);
}